// CubicExpansion_88399016886761
// MI455X (gfx1250) — compile-verified
//
#include <hip/hip_runtime.h>

typedef float v2f __attribute__((ext_vector_type(2)));
typedef float v8f __attribute__((ext_vector_type(8)));

// ---------------------------------------------------------------------------
// Setup kernel: build F (32x32) from knots via Thomas algorithm.
// One wave; lane n owns RHS column n of D (30x32). Faithful to reference:
// B tridiagonal, diag (h[i]+h[i+1])/3, constant off-diag h[30]/6.
// ---------------------------------------------------------------------------
__global__ __launch_bounds__(32) void build_F_kernel(const float* __restrict__ xk,
                                                     float* __restrict__ F) {
  __shared__ float xks[32];
  __shared__ float dp[30][32];   // d' rows for back-substitution
  __shared__ float cp[30];       // c' coefficients
  const int n = threadIdx.x;     // 0..31 column
  xks[n] = xk[n];
  __syncthreads();

  const float e = (xks[31] - xks[30]) * (1.0f / 6.0f);  // off = hs[k-3]/6
  float dprev = 0.0f, cprev = 0.0f;
  for (int i = 0; i < 30; ++i) {
    const float hm = xks[i + 1] - xks[i];
    const float hs = xks[i + 2] - xks[i + 1];
    float Din = 0.0f;
    if (n == i)     Din = 1.0f / hm;
    if (n == i + 1) Din = -1.0f / hm - 1.0f / hs;
    if (n == i + 2) Din = 1.0f / hs;
    const float b  = (hm + hs) * (1.0f / 3.0f);
    const float m  = (i == 0) ? b : (b - e * cprev);
    const float cv = e / m;
    const float dv = (i == 0) ? (Din / m) : ((Din - e * dprev) / m);
    dp[i][n] = dv;
    if (n == 0) cp[i] = cv;
    dprev = dv;
    cprev = cv;
  }
  __syncthreads();

  float fnext = dp[29][n];
  F[30 * 32 + n] = fnext;                 // F row 30 = F_minus row 29
  for (int i = 28; i >= 0; --i) {
    const float fv = dp[i][n] - cp[i] * fnext;
    F[(i + 1) * 32 + n] = fv;
    fnext = fv;
  }
  F[n] = 0.0f;                            // F row 0  = zeros
  F[31 * 32 + n] = 0.0f;                  // F row 31 = zeros
}

// ---------------------------------------------------------------------------
// Main kernel: each wave processes 16 points/tile.
//   base(16x32) = W(16x32) x F(32x32) + E  via 16x V_WMMA_F32_16X16X4_F32
// F is held entirely in 32 VGPRs/wave as WMMA B operands (loaded once).
// E (the a_jm/a_jp identity part) is applied with wave shuffles.
// ---------------------------------------------------------------------------
__global__ __launch_bounds__(256) void cubic_expand_wmma(const float* __restrict__ x,
                                                         const float* __restrict__ F,
                                                         const float* __restrict__ xk,
                                                         float* __restrict__ out,
                                                         int npts) {
  __shared__ float xks[32];
  if (threadIdx.x < 32) xks[threadIdx.x] = xk[threadIdx.x];
  __syncthreads();

  const int lane = threadIdx.x & 31;
  const int wave = threadIdx.x >> 5;
  const int nloc = lane & 15;       // column within 16-wide tile / point index
  const int half = lane >> 4;       // 0: lanes 0-15, 1: lanes 16-31

  // Preload F as WMMA B operands. 32-bit B 4x16 layout (mirrors documented A
  // layout): VGPR0 = K0 (lanes 0-15) / K2 (lanes 16-31); VGPR1 = K1 / K3.
  v2f Bop[2][8];
  #pragma unroll
  for (int t = 0; t < 2; ++t) {
    #pragma unroll
    for (int c = 0; c < 8; ++c) {
      const int k0 = 4 * c + (half ? 2 : 0);
      Bop[t][c].x = F[k0 * 32 + 16 * t + nloc];
      Bop[t][c].y = F[(k0 + 1) * 32 + 16 * t + nloc];
    }
  }

  const float xk0 = xks[0];
  const float invspan = 31.0f / (xks[31] - xk0);

  const int ntiles = npts >> 4;
  const int wave_global = blockIdx.x * (blockDim.x >> 5) + wave;
  const int wstride = gridDim.x * (blockDim.x >> 5);

  for (int tile = wave_global; tile < ntiles; tile += wstride) {
    const int p0 = tile << 4;
    const float xv = x[p0 + nloc];        // both halves load same 16 points

    // searchsorted(xk, x, 'left') via uniform guess + branchless fixup; clip [1,31]
    int g = (int)floorf((xv - xk0) * invspan) + 1;
    g = g < 1 ? 1 : (g > 31 ? 31 : g);
    g = (g < 31 && xks[g] < xv) ? g + 1 : g;
    g = (g > 1 && xks[g - 1] >= xv) ? g - 1 : g;
    const int j = g;

    const float x_j  = xks[j - 1];
    const float x_j1 = xks[j];
    const float h  = x_j1 - x_j;
    const float rh = __builtin_amdgcn_rcpf(h);
    const float dR = x_j1 - xv;
    const float dL = xv - x_j;
    const float a_jm = dR * rh;
    const float a_jp = dL * rh;
    const float c_jm = (dR * dR * rh - h) * dR * (1.0f / 6.0f);
    const float c_jp = (dL * dL * rh - h) * dL * (1.0f / 6.0f);

    // W x F over 8 K-chunks, two 16-col tiles, accumulated in WMMA.
    v8f acc0 = {0.f, 0.f, 0.f, 0.f, 0.f, 0.f, 0.f, 0.f};
    v8f acc1 = {0.f, 0.f, 0.f, 0.f, 0.f, 0.f, 0.f, 0.f};
    const int jm1 = j - 1;
    #pragma unroll
    for (int c = 0; c < 8; ++c) {
      const int k0 = 4 * c + (half ? 2 : 0);   // A layout: VGPR0=K0/K2, VGPR1=K1/K3
      const int k1 = k0 + 1;
      v2f A;
      A.x = (k0 == jm1) ? c_jm : ((k0 == j) ? c_jp : 0.0f);
      A.y = (k1 == jm1) ? c_jm : ((k1 == j) ? c_jp : 0.0f);
      acc0 = __builtin_amdgcn_wmma_f32_16x16x4_f32(false, A, false, Bop[0][c],
                                                   (short)0, acc0, false, false);
      acc1 = __builtin_amdgcn_wmma_f32_16x16x4_f32(false, A, false, Bop[1][c],
                                                   (short)0, acc1, false, false);
    }

    // Identity part (a_jm at col j-1, a_jp at col j) + coalesced b32 stores.
    // D layout: acc[v] holds row (v + 8*half), column nloc (+16 for tile 1).
    #pragma unroll
    for (int v = 0; v < 8; ++v) {
      const int src = v + (half << 3);         // point owned by this D element
      const int   js  = __shfl(j,    src, 32);
      const float ams = __shfl(a_jm, src, 32);
      const float aps = __shfl(a_jp, src, 32);
      const int n0 = nloc;
      const int n1 = nloc + 16;
      const float r0 = acc0[v] + ((js - 1 == n0) ? ams : 0.0f)
                               + ((js     == n0) ? aps : 0.0f);
      const float r1 = acc1[v] + ((js - 1 == n1) ? ams : 0.0f)
                               + ((js     == n1) ? aps : 0.0f);
      const int row = p0 + v + (half << 3);
      out[row * 32 + n0] = r0;   // lanes 0-15 + 16-31: two 64B contiguous runs
      out[row * 32 + n1] = r1;
    }
  }

  // Scalar tail for npts % 16 (npts = 4,000,000 is divisible; kept for safety).
  const int tail = npts & 15;
  if (tail && blockIdx.x == 0 && threadIdx.x < tail) {
    const int p = (npts & ~15) + threadIdx.x;
    const float xv = x[p];
    int g = (int)floorf((xv - xk0) * invspan) + 1;
    g = g < 1 ? 1 : (g > 31 ? 31 : g);
    g = (g < 31 && xks[g] < xv) ? g + 1 : g;
    g = (g > 1 && xks[g - 1] >= xv) ? g - 1 : g;
    const float x_j = xks[g - 1], x_j1 = xks[g];
    const float h = x_j1 - x_j, rh = 1.0f / h;
    const float dR = x_j1 - xv, dL = xv - x_j;
    const float a_jm = dR * rh, a_jp = dL * rh;
    const float c_jm = (dR * dR * rh - h) * dR * (1.0f / 6.0f);
    const float c_jp = (dL * dL * rh - h) * dL * (1.0f / 6.0f);
    for (int c = 0; c < 32; ++c) {
      float val = c_jm * F[(g - 1) * 32 + c] + c_jp * F[g * 32 + c];
      if (c == g - 1) val += a_jm;
      if (c == g)     val += a_jp;
      out[p * 32 + c] = val;
    }
  }
}

// ---------------------------------------------------------------------------
extern "C" void kernel_launch(void* const* d_in, const int* in_sizes, int n_in,
                              void* d_out, int out_size, void* d_ws, size_t ws_size,
                              hipStream_t stream) {
  const float* x  = (const float*)d_in[0];
  const float* xk = (const float*)d_in[1];
  float* out = (float*)d_out;
  float* F   = (float*)d_ws;                 // 32*32 f32 = 4 KB scratch
  const int npts = in_sizes[0];

  build_F_kernel<<<1, 32, 0, stream>>>(xk, F);

  const int ntiles = npts >> 4;
  int blocks = (ntiles + 7) / 8;             // 8 waves/block, 1 tile min each
  if (blocks > 2048) blocks = 2048;          // ~16K waves: plenty of occupancy
  if (blocks < 1) blocks = 1;
  cubic_expand_wmma<<<blocks, 256, 0, stream>>>(x, F, xk, out, npts);
}